// AdditiveAttention_9199819948437
// MI455X (gfx1250) — compile-verified
//
#include <hip/hip_runtime.h>
#include <math.h>

#define HDIM  256
#define BATCH 16
#define QLEN  64
#define KLEN  1024
#define DV    256
#define BQ    (BATCH * QLEN)      // 1024 projected-query rows
#define BKEYS (BATCH * KLEN)      // 16384 projected-key rows
#define NEGBIG -1e6f

typedef float v2f __attribute__((ext_vector_type(2)));
typedef float v8f __attribute__((ext_vector_type(8)));

__device__ __forceinline__ float fast_tanh(float x) {
#if __has_builtin(__builtin_amdgcn_tanhf)
  return __builtin_amdgcn_tanhf(x);
#elif __has_builtin(__builtin_amdgcn_tanh_f32)
  return __builtin_amdgcn_tanh_f32(x);
#else
  return tanhf(x);
#endif
}

// -------------------------------------------------------------------------
// OUT[m, n] = sum_d X[m, d] * W[n, d]      (X: [Mtot, HDIM], W: [HDIM, HDIM])
// One wave per 16x16 output tile; 64 x v_wmma_f32_16x16x4_f32 along d.
// A layout (16x4 f32): lane&15 = M; VGPR v, half = lane>>4 -> K = 2*half + v
// B layout (4x16 f32): lane&15 = N; VGPR v, half -> K = 2*half + v
// D layout (16x16 f32): lane&15 = N; VGPR v, half -> M = 8*half + v
// -------------------------------------------------------------------------
__global__ __launch_bounds__(256) void proj_gemm(const float* __restrict__ X,
                                                 const float* __restrict__ W,
                                                 float* __restrict__ OUT) {
  const int lane = threadIdx.x & 31;
  const int wave = threadIdx.x >> 5;
  const int tile = blockIdx.x * 8 + wave;
  const int ntn  = HDIM / 16;                 // 16 column tiles
  const int tm   = tile / ntn;
  const int tn   = tile % ntn;
  const int half = lane >> 4;
  const int l16  = lane & 15;

  const float* Arow = X + (tm * 16 + l16) * HDIM;         // this lane's A row
  const float* Brow = W + (tn * 16 + l16) * HDIM;         // this lane's B column (W row)

  v8f acc = {};
#pragma unroll 8
  for (int k0 = 0; k0 < HDIM; k0 += 4) {
    v2f a = *(const v2f*)(Arow + k0 + 2 * half);          // K = k0+2h, k0+2h+1
    v2f b = *(const v2f*)(Brow + k0 + 2 * half);
    acc = __builtin_amdgcn_wmma_f32_16x16x4_f32(
        /*neg_a=*/false, a, /*neg_b=*/false, b,
        /*c_mod=*/(short)0, acc, /*reuse_a=*/false, /*reuse_b=*/false);
  }

  float* Orow = OUT + (tm * 16 + 8 * half) * HDIM + tn * 16 + l16;
#pragma unroll
  for (int v = 0; v < 8; ++v)
    Orow[v * HDIM] = acc[v];
}

// -------------------------------------------------------------------------
// Per (b,q): scores[k] = sum_h tanh(Qp[b,q,h] + Kp[b,k,h]) * wv[h], masked
// softmax over k -> attn.
// 256 threads = 8 waves; wave w owns keys [w*128, (w+1)*128).
// Lane l owns hidden slice h = 8l..8l+7 (q/wv held in registers); each key
// row is read as one contiguous, fully-coalesced 1KB wave load, followed by
// 8 hardware-tanh FMAs and a 5-step shfl_xor butterfly reduction.
// -------------------------------------------------------------------------
__global__ __launch_bounds__(256) void score_softmax(const float* __restrict__ Qp,
                                                     const float* __restrict__ Kp,
                                                     const float* __restrict__ wv,
                                                     const int* __restrict__ valid_lens,
                                                     float* __restrict__ attn) {
  __shared__ float scores[KLEN];      // 4 KB
  __shared__ float red[256];

  const int bq   = blockIdx.x;        // 0..1023
  const int b    = bq / QLEN;
  const int t    = threadIdx.x;
  const int lane = t & 31;
  const int wave = t >> 5;

  // Per-lane query / wv slice, resident in registers for all 128 keys.
  const float* qbase = Qp + (size_t)bq * HDIM + lane * 8;
  const float4 q0 = *(const float4*)(qbase);
  const float4 q1 = *(const float4*)(qbase + 4);
  const float* wbase = wv + lane * 8;
  const float4 w0 = *(const float4*)(wbase);
  const float4 w1 = *(const float4*)(wbase + 4);

  const int vlen = valid_lens[b];
  const int kbeg = wave * (KLEN / 8);                     // 128 keys per wave
  const float* krow = Kp + ((size_t)b * KLEN + kbeg) * HDIM + lane * 8;

#pragma unroll 2
  for (int i = 0; i < KLEN / 8; ++i) {
    __builtin_prefetch(krow + 2 * HDIM, 0, 1);            // stream-ahead (speculative)
    const float4 k0v = *(const float4*)(krow);
    const float4 k1v = *(const float4*)(krow + 4);

    float s = fast_tanh(q0.x + k0v.x) * w0.x;
    s += fast_tanh(q0.y + k0v.y) * w0.y;
    s += fast_tanh(q0.z + k0v.z) * w0.z;
    s += fast_tanh(q0.w + k0v.w) * w0.w;
    s += fast_tanh(q1.x + k1v.x) * w1.x;
    s += fast_tanh(q1.y + k1v.y) * w1.y;
    s += fast_tanh(q1.z + k1v.z) * w1.z;
    s += fast_tanh(q1.w + k1v.w) * w1.w;

    // wave32 butterfly: all lanes end with the full dot product
#pragma unroll
    for (int off = 16; off > 0; off >>= 1)
      s += __shfl_xor(s, off, 32);

    const int k = kbeg + i;
    if (lane == 0) scores[k] = (k < vlen) ? s : NEGBIG;
    krow += HDIM;
  }
  __syncthreads();

  // Masked softmax over the 1024 scores (4 values per thread).
  float sc[4];
#pragma unroll
  for (int j = 0; j < 4; ++j) sc[j] = scores[t + 256 * j];

  float m = fmaxf(fmaxf(sc[0], sc[1]), fmaxf(sc[2], sc[3]));
  red[t] = m;
  __syncthreads();
  for (int off = 128; off > 0; off >>= 1) {
    if (t < off) red[t] = fmaxf(red[t], red[t + off]);
    __syncthreads();
  }
  const float M = red[0];
  __syncthreads();

  float p[4];
  float ssum = 0.f;
#pragma unroll
  for (int j = 0; j < 4; ++j) {
    p[j] = __expf(sc[j] - M);
    ssum += p[j];
  }
  red[t] = ssum;
  __syncthreads();
  for (int off = 128; off > 0; off >>= 1) {
    if (t < off) red[t] += red[t + off];
    __syncthreads();
  }
  const float inv = 1.f / red[0];

#pragma unroll
  for (int j = 0; j < 4; ++j)
    attn[(size_t)bq * KLEN + t + 256 * j] = p[j] * inv;
}

// -------------------------------------------------------------------------
// OUT[b] (64x256) = attn[b] (64x1024) @ values[b] (1024x256), f32 WMMA.
// -------------------------------------------------------------------------
__global__ __launch_bounds__(256) void out_gemm(const float* __restrict__ attn,
                                                const float* __restrict__ V,
                                                float* __restrict__ OUT) {
  const int lane = threadIdx.x & 31;
  const int wave = threadIdx.x >> 5;
  const int tile = blockIdx.x * 8 + wave;     // 16 batches * 4 * 16 = 1024 tiles
  const int ntn  = DV / 16;                   // 16
  const int ntm  = QLEN / 16;                 // 4
  const int b    = tile / (ntm * ntn);
  const int r    = tile % (ntm * ntn);
  const int tm   = r / ntn;
  const int tn   = r % ntn;
  const int half = lane >> 4;
  const int l16  = lane & 15;

  const float* Ab = attn + (size_t)b * QLEN * KLEN + (tm * 16 + l16) * KLEN;
  const float* Vb = V + (size_t)b * KLEN * DV + tn * 16 + l16;

  v8f acc = {};
#pragma unroll 4
  for (int k0 = 0; k0 < KLEN; k0 += 4) {
    const int kb = k0 + 2 * half;
    v2f a = *(const v2f*)(Ab + kb);            // consecutive along k
    v2f bv;
    bv.x = Vb[(size_t)kb * DV];                // row kb,   col n
    bv.y = Vb[(size_t)(kb + 1) * DV];          // row kb+1, col n
    acc = __builtin_amdgcn_wmma_f32_16x16x4_f32(
        false, a, false, bv, (short)0, acc, false, false);
  }

  float* Orow = OUT + (size_t)(b * QLEN + tm * 16 + 8 * half) * DV + tn * 16 + l16;
#pragma unroll
  for (int v = 0; v < 8; ++v)
    Orow[v * DV] = acc[v];
}

// -------------------------------------------------------------------------
extern "C" void kernel_launch(void* const* d_in, const int* in_sizes, int n_in,
                              void* d_out, int out_size, void* d_ws, size_t ws_size,
                              hipStream_t stream) {
  const float* queries = (const float*)d_in[0];  // [16,64,256]
  const float* keys    = (const float*)d_in[1];  // [16,1024,256]
  const float* values  = (const float*)d_in[2];  // [16,1024,256]
  const int*   vlens   = (const int*)d_in[3];    // [16]
  const float* Wq      = (const float*)d_in[4];  // [256,256]
  const float* Wk      = (const float*)d_in[5];  // [256,256]
  const float* wv      = (const float*)d_in[6];  // [256]
  float* out = (float*)d_out;                    // [16,64,256]

  float* Qp   = (float*)d_ws;                    // 1024*256   floats ( 1 MB)
  float* Kp   = Qp + (size_t)BQ * HDIM;          // 16384*256  floats (16 MB)
  float* attn = Kp + (size_t)BKEYS * HDIM;       // 1024*1024  floats ( 4 MB)

  // Projections: one wave per 16x16 tile, 8 waves per block.
  proj_gemm<<<(BQ / 16) * (HDIM / 16) / 8, 256, 0, stream>>>(queries, Wq, Qp);
  proj_gemm<<<(BKEYS / 16) * (HDIM / 16) / 8, 256, 0, stream>>>(keys, Wk, Kp);

  // Additive scores + masked softmax: one block per (b,q).
  score_softmax<<<BQ, 256, 0, stream>>>(Qp, Kp, wv, vlens, attn);

  // attn @ values.
  out_gemm<<<(BATCH * (QLEN / 16) * (DV / 16)) / 8, 256, 0, stream>>>(attn, values, out);
}